// RenderingKaolin_22497038696984
// MI455X (gfx1250) — compile-verified
//
#include <hip/hip_runtime.h>
#include <math.h>
#include <stdint.h>

typedef float v2f __attribute__((ext_vector_type(2)));
typedef float v8f __attribute__((ext_vector_type(8)));

#define HEIGHT 128
#define WIDTH 128
#define NPIX 16384
#define NV 600
#define NF 1000
#define NFPAD 1008
#define NGRP 63
#define CAM_DIST 2.0f
#define TEXW 512
#define REC_STRIDE 16

// -------------------- kernel 0: rotate + translate vertices --------------------
__global__ void k_verts(const float* __restrict__ q, const float* __restrict__ tr,
                        const float* __restrict__ uv, float* __restrict__ wsverts) {
  int v = blockIdx.x * 256 + threadIdx.x;
  if (v >= NV) return;
  float u0 = uv[v*3+0], u1 = uv[v*3+1], u2 = uv[v*3+2];
  for (int frm = 0; frm < 2; ++frm) {
    float w = q[frm*4+0], x = q[frm*4+1], y = q[frm*4+2], z = q[frm*4+3];
    float inv = 1.f / sqrtf(w*w + x*x + y*y + z*z);
    w *= inv; x *= inv; y *= inv; z *= inv;
    float R00 = 1.f-2.f*(y*y+z*z), R01 = 2.f*(x*y-w*z),     R02 = 2.f*(x*z+w*y);
    float R10 = 2.f*(x*y+w*z),     R11 = 1.f-2.f*(x*x+z*z), R12 = 2.f*(y*z-w*x);
    float R20 = 2.f*(x*z-w*y),     R21 = 2.f*(y*z+w*x),     R22 = 1.f-2.f*(x*x+y*y);
    float* o = wsverts + ((size_t)frm*NV + v)*3;
    o[0] = R00*u0 + R01*u1 + R02*u2 + tr[0*2+frm];
    o[1] = R10*u0 + R11*u1 + R12*u2 + tr[1*2+frm];
    o[2] = R20*u0 + R21*u1 + R22*u2 + tr[2*2+frm];
  }
}

// ---- kernel 1: per-face setup (edge coeffs prescaled by inv_area) + flow out ----
__global__ void k_faces(const int* __restrict__ faces, const float* __restrict__ wsverts,
                        float* __restrict__ wsrec, float* __restrict__ flow) {
  int f = blockIdx.x * 256 + threadIdx.x;
  if (f >= NFPAD) return;
  if (f >= NF) {  // padding faces: always-invalid record
    for (int frm = 0; frm < 2; ++frm) {
      float* rp = wsrec + ((size_t)frm*NFPAD + f)*REC_STRIDE;
      #pragma unroll
      for (int t = 0; t < 16; ++t) rp[t] = 0.f;
      rp[2] = -1.f; rp[6] = -1.f; rp[10] = -1.f;  // bary consts < 0, nz (rp[11]) = 0
    }
    return;
  }
  const float FFAC = 1.f / tanf(0.5f * (1.57f * 0.5f));
  int idxs[3] = { faces[f*3+0], faces[f*3+1], faces[f*3+2] };
  float fimg[2][3][2];
  for (int frm = 0; frm < 2; ++frm) {
    float fc[3][3];
    for (int j = 0; j < 3; ++j) {
      const float* vp = wsverts + ((size_t)frm*NV + idxs[j])*3;
      fc[j][0] = vp[0]; fc[j][1] = vp[1]; fc[j][2] = vp[2] - CAM_DIST;
      float invz = 1.f / (-fc[j][2]);
      fimg[frm][j][0] = fc[j][0] * FFAC * invz;
      fimg[frm][j][1] = fc[j][1] * FFAC * invz;
    }
    float e1x = fc[1][0]-fc[0][0], e1y = fc[1][1]-fc[0][1], e1z = fc[1][2]-fc[0][2];
    float e2x = fc[2][0]-fc[0][0], e2y = fc[2][1]-fc[0][1], e2z = fc[2][2]-fc[0][2];
    float nx = e1y*e2z - e1z*e2y, ny = e1z*e2x - e1x*e2z, nz = e1x*e2y - e1y*e2x;
    float nrm = sqrtf(nx*nx + ny*ny + nz*nz) + 1e-10f;
    nx /= nrm; ny /= nrm; nz /= nrm;
    float ax = fimg[frm][0][0], ay = fimg[frm][0][1];
    float bx = fimg[frm][1][0], by = fimg[frm][1][1];
    float cx = fimg[frm][2][0], cy = fimg[frm][2][1];
    float area = (bx-ax)*(cy-ay) - (by-ay)*(cx-ax);
    float ex[3], ey[3], qx[3], qy[3];
    ex[0] = cx-bx; ey[0] = cy-by; qx[0] = bx; qy[0] = by;
    ex[1] = ax-cx; ey[1] = ay-cy; qx[1] = cx; qy[1] = cy;
    ex[2] = bx-ax; ey[2] = by-ay; qx[2] = ax; qy[2] = ay;
    float den = (fabsf(area) < 1e-10f) ? 1e-10f : area;
    float ia = 1.f / den;
    float* rp = wsrec + ((size_t)frm*NFPAD + f)*REC_STRIDE;
    float fz[3] = { fc[0][2], fc[1][2], fc[2][2] };
    float zc0 = 0.f, zc1 = 0.f, zc2 = 0.f;
    #pragma unroll
    for (int i = 0; i < 3; ++i) {
      float c0 = ex[i] * ia;                              // coeff of P.y
      float c1 = -ey[i] * ia;                             // coeff of P.x
      float c2 = (ey[i]*qx[i] - ex[i]*qy[i]) * ia;        // const
      rp[i*4+0] = c0; rp[i*4+1] = c1; rp[i*4+2] = c2;
      zc0 += fz[i]*c0; zc1 += fz[i]*c1; zc2 += fz[i]*c2;
    }
    rp[3] = nx; rp[7] = ny; rp[11] = nz; rp[15] = 0.f;
    rp[12] = zc0; rp[13] = zc1; rp[14] = zc2;
  }
  for (int j = 0; j < 3; ++j)
    for (int k = 0; k < 2; ++k)
      flow[(f*3+j)*2 + k] = fimg[1][j][k] - fimg[0][j][k];
}

// ------ kernel 2: WMMA rasterizer + shading (one image row per 256-thr block) ------
__global__ __launch_bounds__(256)
void k_raster(const float* __restrict__ wsrec, const float* __restrict__ ff,
              const float* __restrict__ tex, const float* __restrict__ lights,
              float* __restrict__ outc, float* __restrict__ wsmask) {
  __shared__ float4 recv4[NFPAD * REC_STRIDE / 4];
  float* rec = (float*)recv4;
  const int tid = threadIdx.x;
  const int frm = blockIdx.x >> 7;
  const int row = blockIdx.x & 127;

  if (tid == 0) __builtin_prefetch(tex, 0, 3);  // warm texture path (global_prefetch_b8)

  // Stage 63KB of face records into LDS via CDNA5 async global->LDS DMA
  // (GLOBAL_LOAD_ASYNC_TO_LDS_B128, GV mode: vdst = LDS byte addr, vaddr = 64-bit
  //  global addr; tracked by ASYNCcnt, no VGPR staging).
  {
    const float4* g4 = (const float4*)(wsrec + (size_t)frm * NFPAD * REC_STRIDE);
    // Flat shared pointer truncates to the workgroup-relative LDS byte address.
    uint32_t lds0 = (uint32_t)(uintptr_t)(&recv4[0]);
    for (int i = tid; i < NFPAD*REC_STRIDE/4; i += 256) {
      uint32_t ldsa = lds0 + (uint32_t)i * 16u;
      uint64_t ga   = (uint64_t)(uintptr_t)(g4 + i);
      asm volatile("global_load_async_to_lds_b128 %0, %1, off"
                   :: "v"(ldsa), "v"(ga)
                   : "memory");
    }
    asm volatile("s_wait_asynccnt 0x0" ::: "memory");
  }
  __syncthreads();

  const int wv   = tid >> 5;
  const int lane = tid & 31;
  const int half = lane >> 4;
  const int ln   = lane & 15;

  const float py  = 1.f - (row + 0.5f) * (2.f / HEIGHT);
  const float pxl = ((wv*16 + ln) + 0.5f) * (2.f / WIDTH) - 1.f;
  // A (16x4 f32): lanes 0-15 hold K0,K1 = (P.y, P.x); lanes 16-31 hold K2,K3 = (1, 0)
  v2f A;
  A.x = half ? 1.f : py;
  A.y = half ? 0.f : pxl;

  float best[8]; int bestf[8];
  #pragma unroll
  for (int r = 0; r < 8; ++r) { best[r] = -1e10f; bestf[r] = 0; }

  for (int g = 0; g < NGRP; ++g) {
    const int f = g*16 + ln;
    const float* rp = rec + f*REC_STRIDE;
    // B (4x16 f32): lanes 0-15 hold K0,K1; lanes 16-31 hold K2,K3 (column = lane%16)
    v2f B0, B1, B2, BZ;
    if (half == 0) {
      B0.x = rp[0];  B0.y = rp[1];
      B1.x = rp[4];  B1.y = rp[5];
      B2.x = rp[8];  B2.y = rp[9];
      BZ.x = rp[12]; BZ.y = rp[13];
    } else {
      B0.x = rp[2];  B0.y = 0.f;
      B1.x = rp[6];  B1.y = 0.f;
      B2.x = rp[10]; B2.y = 0.f;
      BZ.x = rp[14]; BZ.y = 0.f;
    }
    const float nzf = rp[11];
    v8f Z = {0.f,0.f,0.f,0.f,0.f,0.f,0.f,0.f};
    v8f d0 = __builtin_amdgcn_wmma_f32_16x16x4_f32(false, A, false, B0, (short)0, Z, false, false);
    v8f d1 = __builtin_amdgcn_wmma_f32_16x16x4_f32(false, A, false, B1, (short)0, Z, false, false);
    v8f d2 = __builtin_amdgcn_wmma_f32_16x16x4_f32(false, A, false, B2, (short)0, Z, false, false);
    v8f dz = __builtin_amdgcn_wmma_f32_16x16x4_f32(false, A, false, BZ, (short)0, Z, false, false);
    #pragma unroll
    for (int r = 0; r < 8; ++r) {
      bool valid = (d0[r] >= 0.f) && (d1[r] >= 0.f) && (d2[r] >= 0.f) && (nzf > 0.f);
      float sc = valid ? dz[r] : -1e10f;
      if (sc > best[r]) { best[r] = sc; bestf[r] = f; }
    }
  }

  // argmax across the 16 face-column lanes of each half (lowest index on ties)
  #pragma unroll
  for (int m = 1; m < 16; m <<= 1) {
    #pragma unroll
    for (int r = 0; r < 8; ++r) {
      float os = __shfl_xor(best[r],  m, 32);
      int   of = __shfl_xor(bestf[r], m, 32);
      if (os > best[r] || (os == best[r] && of < bestf[r])) { best[r] = os; bestf[r] = of; }
    }
  }

  if (ln < 8) {  // 16 active lanes per wave: one pixel each
    const int r   = ln;
    const int M   = r + half*8;          // D-matrix row -> pixel index in tile
    const int x   = wv*16 + M;
    const int y   = row;
    const int red = bestf[r];
    const bool has = best[r] > -1e9f;
    const float ppx = (x + 0.5f) * (2.f / WIDTH) - 1.f;
    const float ppy = py;
    const float* rp = rec + red*REC_STRIDE;
    float u = 0.f, vv = 0.f;
    #pragma unroll
    for (int i = 0; i < 3; ++i) {
      float bi = ppy*rp[i*4+0] + ppx*rp[i*4+1] + rp[i*4+2];   // bary (prescaled)
      u  += bi * ff[(red*3+i)*2+0];
      vv += bi * ff[(red*3+i)*2+1];
    }
    if (!has) { u = 0.f; vv = 0.f; }
    float tx = fminf(fmaxf(u, 0.f), 1.f) * (float)(TEXW-1);
    float ty = (1.f - fminf(fmaxf(vv, 0.f), 1.f)) * (float)(TEXW-1);
    float x0f = floorf(tx), y0f = floorf(ty);
    float wx = tx - x0f, wy = ty - y0f;
    int x0 = (int)fminf(fmaxf(x0f, 0.f), (float)(TEXW-1));
    int x1 = (int)fminf(x0f + 1.f, (float)(TEXW-1));
    int y0 = (int)fminf(fmaxf(y0f, 0.f), (float)(TEXW-1));
    int y1 = (int)fminf(y0f + 1.f, (float)(TEXW-1));
    float nx = rp[3], ny = rp[7], nzv = rp[11];
    float basis[9];
    basis[0] = 0.282095f;
    basis[1] = 0.488603f * ny;
    basis[2] = 0.488603f * nzv;
    basis[3] = 0.488603f * nx;
    basis[4] = 1.092548f * nx * ny;
    basis[5] = 1.092548f * ny * nzv;
    basis[6] = 0.315392f * (3.f*nzv*nzv - 1.f);
    basis[7] = 1.092548f * nx * nzv;
    basis[8] = 0.546274f * (nx*nx - ny*ny);
    const float w00 = (1.f-wx)*(1.f-wy), w01 = wx*(1.f-wy), w10 = (1.f-wx)*wy, w11 = wx*wy;
    const int o00 = y0*TEXW+x0, o01 = y0*TEXW+x1, o10 = y1*TEXW+x0, o11 = y1*TEXW+x1;
    float* outf = outc + (size_t)frm * 4 * NPIX;
    #pragma unroll
    for (int c = 0; c < 3; ++c) {
      const float* tc = tex + (size_t)c * TEXW * TEXW;
      float s = tc[o00]*w00 + tc[o01]*w01 + tc[o10]*w10 + tc[o11]*w11;
      float L = 1.f;
      if (has) {
        L = 0.f;
        #pragma unroll
        for (int k = 0; k < 9; ++k) L += basis[k] * lights[c*9+k];
      }
      outf[c*NPIX + y*WIDTH + x] = s * L;
    }
    wsmask[frm*NPIX + y*WIDTH + x] = has ? 1.f : 0.f;
  }
}

// -------------------- kernel 3: 3x3 min-erode of the mask --------------------
__global__ void k_erode(const float* __restrict__ wsmask, float* __restrict__ outc) {
  int p = blockIdx.x * 256 + threadIdx.x;
  if (p >= 2*NPIX) return;
  int frm = p / NPIX, idx = p % NPIX;
  int y = idx >> 7, x = idx & 127;
  const float* mk = wsmask + frm*NPIX;
  float mn = 10000.f;
  for (int dy = -1; dy <= 1; ++dy) {
    int yy = y + dy; if (yy < 0 || yy >= HEIGHT) continue;
    for (int dx = -1; dx <= 1; ++dx) {
      int xx = x + dx; if (xx < 0 || xx >= WIDTH) continue;
      mn = fminf(mn, mk[yy*WIDTH + xx]);
    }
  }
  outc[(size_t)frm*4*NPIX + 3*NPIX + idx] = mn;
}

extern "C" void kernel_launch(void* const* d_in, const int* in_sizes, int n_in,
                              void* d_out, int out_size, void* d_ws, size_t ws_size,
                              hipStream_t stream) {
  (void)in_sizes; (void)n_in; (void)out_size; (void)ws_size;
  const float* translation   = (const float*)d_in[0];
  const float* quaternion    = (const float*)d_in[1];
  const float* unit_vertices = (const float*)d_in[2];
  const float* face_features = (const float*)d_in[3];
  const float* texture       = (const float*)d_in[4];
  const float* lights        = (const float*)d_in[5];
  const int*   faces         = (const int*)d_in[6];
  float* out = (float*)d_out;
  float* ws  = (float*)d_ws;
  float* wsverts = ws;                    // 2*600*3   = 3600 floats
  float* wsrec   = ws + 3600;             // 2*1008*16 = 32256 floats
  float* wsmask  = ws + 3600 + 32256;     // 2*16384   = 32768 floats
  float* flowout = out + 2*4*NPIX;        // flow tail of d_out (6000 floats)

  hipLaunchKernelGGL(k_verts,  dim3((NV+255)/256),    dim3(256), 0, stream,
                     quaternion, translation, unit_vertices, wsverts);
  hipLaunchKernelGGL(k_faces,  dim3((NFPAD+255)/256), dim3(256), 0, stream,
                     faces, wsverts, wsrec, flowout);
  hipLaunchKernelGGL(k_raster, dim3(256),             dim3(256), 0, stream,
                     wsrec, face_features, texture, lights, out, wsmask);
  hipLaunchKernelGGL(k_erode,  dim3(128),             dim3(256), 0, stream,
                     wsmask, out);
}